// MultiVarAttentionROM_49503793054027
// MI455X (gfx1250) — compile-verified
//
#include <hip/hip_runtime.h>
#include <hip/hip_bf16.h>
#include <math.h>

// ---------------------------------------------------------------------------
// Problem constants (match reference)
// ---------------------------------------------------------------------------
#define Bq   512
#define Vv   2
#define NXq  1024
#define Rq   64
#define Dq   512
#define Sq   4
#define HSq  64
#define FINq 384   // S*HS + 128

typedef float v2f __attribute__((ext_vector_type(2)));
typedef float v4f __attribute__((ext_vector_type(4)));
typedef float v8f __attribute__((ext_vector_type(8)));

__device__ __forceinline__ float gelu_t(float x) {
    // jax.nn.gelu default (tanh approximation)
    float x3 = x * x * x;
    float t  = tanhf(0.7978845608028654f * (x + 0.044715f * x3));
    return 0.5f * x * (1.0f + t);
}

// ---------------------------------------------------------------------------
// Generic strided f32 WMMA GEMM:
//   C[m, n] = epilogue( sum_k A[m*sa_m + k*sa_k] * W[k*sw_k + n*sw_n] + bias[n] )
// REQUIREMENT (guaranteed by all call sites): M % 32 == 0, N % 64 == 0,
// K % 32 == 0 -> no bounds guards anywhere, EXEC stays all-ones (WMMA req).
// Template flags A_UNIT / W_UNIT select branch-free vectorized staging
// (b128 global loads) vs strided gather (Phi^T call sites).
// Block: 256 threads = 8 waves; block tile 32(M) x 64(N); K chunk = 32.
// Each wave owns one 16x16 C tile via V_WMMA_F32_16X16X4_F32 (f32-exact).
// act: 0=none, 1=gelu, 2=relu.  acc_flag: 1 -> C[idx] += val (U_B residual).
// ---------------------------------------------------------------------------
template <bool A_UNIT, bool W_UNIT>
__global__ __launch_bounds__(256) void gemm_wmma_f32(
    const float* __restrict__ A, const float* __restrict__ W,
    const float* __restrict__ bias, float* __restrict__ C,
    int M, int N, int K,
    int sa_m, int sa_k, int sw_k, int sw_n,
    int ldc, int c_off, int act, int acc_flag)
{
    __shared__ float As[32][36];   // [m][k], pad 36 -> conflict-free col reads
    __shared__ float Ws[32][68];   // [k][n], pad 68 -> b128-friendly + no conflicts

    const int t       = threadIdx.x;
    const int lane    = t & 31;
    const int wid     = t >> 5;
    const int waveM   = wid >> 2;        // 0..1
    const int waveN   = wid & 3;         // 0..3
    const int mBase   = blockIdx.y * 32;
    const int nBase   = blockIdx.x * 64;
    const int halfsel = (lane >> 4) & 1; // lanes 16..31 take K+2/K+3
    const int mloc    = waveM * 16 + (lane & 15);
    const int nloc    = waveN * 16 + (lane & 15);

    // staging maps (computed once)
    const int am  = t >> 3;              // 0..31  (A row)
    const int ak4 = (t & 7) * 4;         // 0,4,..28 (A col base, float4)
    const int wk  = t >> 3;              // 0..31  (W row)
    const int wn8 = (t & 7) * 8;         // 0,8,..56 (W col base, 2x float4)

    v8f acc = {0.f, 0.f, 0.f, 0.f, 0.f, 0.f, 0.f, 0.f};

    for (int k0 = 0; k0 < K; k0 += 32) {
        // ---- stage A tile (32x32) ----
        if (A_UNIT) {
            const float* src = A + (long)(mBase + am) * sa_m + (k0 + ak4);
            v4f x = *(const v4f*)src;
            As[am][ak4 + 0] = x.x; As[am][ak4 + 1] = x.y;
            As[am][ak4 + 2] = x.z; As[am][ak4 + 3] = x.w;
        } else {
            // strided (Phi^T): 4 elements per thread, unconditional
            const float* src = A + (long)(mBase + am) * sa_m
                                 + (long)(k0 + ak4) * sa_k;
#pragma unroll
            for (int j = 0; j < 4; j++)
                As[am][ak4 + j] = src[(long)j * sa_k];
        }
        // ---- stage W tile (32x64) ----
        if (W_UNIT) {
            const float* src = W + (long)(k0 + wk) * sw_k + (nBase + wn8);
            v4f x0 = *(const v4f*)src;
            v4f x1 = *(const v4f*)(src + 4);
            Ws[wk][wn8 + 0] = x0.x; Ws[wk][wn8 + 1] = x0.y;
            Ws[wk][wn8 + 2] = x0.z; Ws[wk][wn8 + 3] = x0.w;
            Ws[wk][wn8 + 4] = x1.x; Ws[wk][wn8 + 5] = x1.y;
            Ws[wk][wn8 + 6] = x1.z; Ws[wk][wn8 + 7] = x1.w;
        } else {
            const float* src = W + (long)(k0 + wk) * sw_k
                                 + (long)(nBase + wn8) * sw_n;
#pragma unroll
            for (int j = 0; j < 8; j++)
                Ws[wk][wn8 + j] = src[(long)j * sw_n];
        }
        __syncthreads();

#pragma unroll
        for (int kk = 0; kk < 32; kk += 4) {
            const int kb = kk + halfsel * 2;
            // A 16x4 layout: lanes0-15 {K=kk,kk+1}, lanes16-31 {K=kk+2,kk+3}
            v2f a;  a.x  = As[mloc][kb];  a.y  = As[mloc][kb + 1];
            // B 4x16 layout: rows striped across lanes within a VGPR
            v2f bf; bf.x = Ws[kb][nloc];  bf.y = Ws[kb + 1][nloc];
            acc = __builtin_amdgcn_wmma_f32_16x16x4_f32(
                      false, a, false, bf, (short)0, acc, false, false);
        }
        __syncthreads();
    }

    // Epilogue: C/D layout -> VGPR j: lanes0-15 M=j, lanes16-31 M=j+8
    const int n     = nBase + waveN * 16 + (lane & 15);
    const int mrow0 = mBase + waveM * 16 + halfsel * 8;
    const float bv  = (bias != nullptr) ? bias[n] : 0.0f;
#pragma unroll
    for (int j = 0; j < 8; j++) {
        const int m = mrow0 + j;
        float val = acc[j] + bv;
        if (act == 1)      val = gelu_t(val);
        else if (act == 2) val = fmaxf(val, 0.0f);
        const long idx = (long)m * ldc + n + c_off;
        if (acc_flag) val += C[idx];
        C[idx] = val;
    }
}

// ---------------------------------------------------------------------------
// Control-feature builder: f[b, 0:256] = gelu(bs ⊗ sb_w + sb_b),
//                          f[b, 256:384] = gelu(gelu(bc@cm_w1+b1)@cm_w2+b2)
// ---------------------------------------------------------------------------
__global__ __launch_bounds__(256) void build_f_kernel(
    const float* __restrict__ BC, const float* __restrict__ sb_w,
    const float* __restrict__ sb_b, const float* __restrict__ cm_w1,
    const float* __restrict__ cm_b1, const float* __restrict__ cm_w2,
    const float* __restrict__ cm_b2, float* __restrict__ f)
{
    const int b = blockIdx.x;
    const int t = threadIdx.x;
    __shared__ float c1s[64];

    const float bc4 = BC[b * 6 + 4];
    const float bc5 = BC[b * 6 + 5];
    if (t < 64)
        c1s[t] = gelu_t(bc4 * cm_w1[t] + bc5 * cm_w1[64 + t] + cm_b1[t]);

    {   // structure features: S=4 heads of HS=64, sb_w is (S,1,HS)
        int s = t >> 6, j = t & 63;
        float bs = BC[b * 6 + s];
        f[b * FINq + t] = gelu_t(bs * sb_w[s * 64 + j] + sb_b[s * 64 + j]);
    }
    __syncthreads();
    if (t < 128) {
        float a = cm_b2[t];
        for (int i = 0; i < 64; i++) a += c1s[i] * cm_w2[i * 128 + t];
        f[b * FINq + 256 + t] = gelu_t(a);
    }
}

// ---------------------------------------------------------------------------
// Per-v attention collapse: C1[v], C0[v].
//   qk[r,h]  = sum_k Wq_w[v,h*64+k] * K[v,r,h*64+k]   (qb with Wq_b)
//   Wkv[d]   = sum_r qk[r,h(d)] * Vm[v,r,d]           (Bkv with qb)
//   t1[e]    = sum_d Wkv[d] * attn_ow[v,d,e]          (t0 + attn_ob)
//   C1[v]    = (t1 . proj_w)/sqrt(D)
//   C0[v]    = ((t0+attn_ob) . proj_w)/sqrt(D) + proj_b[v]
// ---------------------------------------------------------------------------
__global__ __launch_bounds__(512) void consts_kernel(
    const float* __restrict__ Kb, const float* __restrict__ Vb,
    const float* __restrict__ Wq_w, const float* __restrict__ Wq_b,
    const float* __restrict__ attn_ow, const float* __restrict__ attn_ob,
    const float* __restrict__ proj_w, const float* __restrict__ proj_b,
    float* __restrict__ Cc)
{
    const int v = blockIdx.x;
    const int t = threadIdx.x;   // 512
    __shared__ float qk[512], qb[512], Wkv[512], Bkv[512], red[512];

    const float* Kv = Kb + v * 64 * 512;
    const float* Vm = Vb + v * 64 * 512;

    {   // qk / qb
        int r = t >> 3, h = t & 7;
        const float* qw  = Wq_w + v * 512 + h * 64;
        const float* qbv = Wq_b + v * 512 + h * 64;
        const float* kr  = Kv + r * 512 + h * 64;
        float s1 = 0.f, s0 = 0.f;
        for (int k = 0; k < 64; k++) { s1 += qw[k] * kr[k]; s0 += qbv[k] * kr[k]; }
        qk[r * 8 + h] = s1; qb[r * 8 + h] = s0;
    }
    __syncthreads();
    {   // Wkv / Bkv
        int d = t, h = d >> 6;
        float s1 = 0.f, s0 = 0.f;
        for (int r = 0; r < 64; r++) {
            float vv = Vm[r * 512 + d];
            s1 += qk[r * 8 + h] * vv;
            s0 += qb[r * 8 + h] * vv;
        }
        Wkv[d] = s1; Bkv[d] = s0;
    }
    __syncthreads();
    float t1 = 0.f, t0 = 0.f;
    {
        const float* ow = attn_ow + v * 512 * 512 + t;  // column e=t, stride D over d
        for (int d = 0; d < 512; d++) {
            float w = ow[(long)d * 512];
            t1 += Wkv[d] * w;
            t0 += Bkv[d] * w;
        }
        t0 += attn_ob[v * 512 + t];
    }
    const float pw = proj_w[v * 512 + t];
    const float inv_sqrtD = 0.04419417382415922f;   // 1/sqrt(512)

    red[t] = t1 * pw; __syncthreads();
    for (int s = 256; s > 0; s >>= 1) { if (t < s) red[t] += red[t + s]; __syncthreads(); }
    if (t == 0) Cc[v * 2 + 0] = red[0] * inv_sqrtD;
    __syncthreads();
    red[t] = t0 * pw; __syncthreads();
    for (int s = 256; s > 0; s >>= 1) { if (t < s) red[t] += red[t + s]; __syncthreads(); }
    if (t == 0) Cc[v * 2 + 1] = red[0] * inv_sqrtD + proj_b[v];
}

// ---------------------------------------------------------------------------
// Row LayerNorm over 64 elements (a_map = LN(relu-GEMM output))
// ---------------------------------------------------------------------------
__global__ __launch_bounds__(64) void ln64_kernel(
    const float* __restrict__ X, const float* __restrict__ g,
    const float* __restrict__ bt, float* __restrict__ OUT)
{
    const int b = blockIdx.x, v = blockIdx.y, t = threadIdx.x;
    __shared__ float s[64];
    const int base = v * Bq * 64 + b * 64;
    float x = X[base + t];
    s[t] = x; __syncthreads();
    for (int st = 32; st > 0; st >>= 1) { if (t < st) s[t] += s[t + st]; __syncthreads(); }
    float mu = s[0] * (1.0f / 64.0f);
    __syncthreads();
    float dx = x - mu;
    s[t] = dx * dx; __syncthreads();
    for (int st = 32; st > 0; st >>= 1) { if (t < st) s[t] += s[t + st]; __syncthreads(); }
    float var = s[0] * (1.0f / 64.0f);
    OUT[base + t] = dx * rsqrtf(var + 1e-5f) * g[v * 64 + t] + bt[v * 64 + t];
}

// ---------------------------------------------------------------------------
// ffn_update = LN(O + a_map); a_next = (1+C1)*a_map + C0 + alpha*ffn_update
// ---------------------------------------------------------------------------
__global__ __launch_bounds__(64) void anext_kernel(
    const float* __restrict__ O, const float* __restrict__ amap,
    const float* __restrict__ g, const float* __restrict__ bt,
    const float* __restrict__ Cc, const float* __restrict__ alpha,
    float* __restrict__ anext)
{
    const int b = blockIdx.x, v = blockIdx.y, t = threadIdx.x;
    __shared__ float s[64];
    const int base = v * Bq * 64 + b * 64;
    float am = amap[base + t];
    float x  = O[base + t] + am;
    s[t] = x; __syncthreads();
    for (int st = 32; st > 0; st >>= 1) { if (t < st) s[t] += s[t + st]; __syncthreads(); }
    float mu = s[0] * (1.0f / 64.0f);
    __syncthreads();
    float dx = x - mu;
    s[t] = dx * dx; __syncthreads();
    for (int st = 32; st > 0; st >>= 1) { if (t < st) s[t] += s[t + st]; __syncthreads(); }
    float var = s[0] * (1.0f / 64.0f);
    float u = dx * rsqrtf(var + 1e-5f) * g[v * 64 + t] + bt[v * 64 + t];
    float C1 = Cc[v * 2 + 0], C0 = Cc[v * 2 + 1];
    anext[base + t] = (1.0f + C1) * am + C0 + alpha[v] * u;
}

// ---------------------------------------------------------------------------
// Host-side orchestration
// ---------------------------------------------------------------------------
static inline dim3 gemm_grid(int M, int N) { return dim3(N / 64, M / 32); }

extern "C" void kernel_launch(void* const* d_in, const int* in_sizes, int n_in,
                              void* d_out, int out_size, void* d_ws, size_t ws_size,
                              hipStream_t stream) {
    (void)in_sizes; (void)n_in; (void)out_size; (void)ws_size;
    const float* a_n      = (const float*)d_in[0];
    const float* BC       = (const float*)d_in[1];
    const float* Phi      = (const float*)d_in[2];
    const float* Wq_w     = (const float*)d_in[3];
    const float* Wq_b     = (const float*)d_in[4];
    const float* Wk_w     = (const float*)d_in[5];
    const float* Wk_b     = (const float*)d_in[6];
    const float* Wv_w     = (const float*)d_in[7];
    const float* Wv_b     = (const float*)d_in[8];
    const float* attn_ow  = (const float*)d_in[9];
    const float* attn_ob  = (const float*)d_in[10];
    const float* proj_w   = (const float*)d_in[11];
    const float* proj_b   = (const float*)d_in[12];
    const float* ffn_w1   = (const float*)d_in[13];
    const float* ffn_b1   = (const float*)d_in[14];
    const float* ffn_w2   = (const float*)d_in[15];
    const float* ffn_b2   = (const float*)d_in[16];
    const float* ffn_w3   = (const float*)d_in[17];
    const float* ffn_b3   = (const float*)d_in[18];
    const float* ffn_ln_g = (const float*)d_in[19];
    const float* ffn_ln_b = (const float*)d_in[20];
    const float* a0_w     = (const float*)d_in[21];
    const float* a0_b     = (const float*)d_in[22];
    const float* a0_ln_g  = (const float*)d_in[23];
    const float* a0_ln_b  = (const float*)d_in[24];
    const float* alpha    = (const float*)d_in[25];
    const float* sb_w     = (const float*)d_in[26];
    const float* sb_b     = (const float*)d_in[27];
    const float* cm_w1    = (const float*)d_in[28];
    const float* cm_b1    = (const float*)d_in[29];
    const float* cm_w2    = (const float*)d_in[30];
    const float* cm_b2    = (const float*)d_in[31];
    const float* fu_w1    = (const float*)d_in[32];
    const float* fu_b1    = (const float*)d_in[33];
    const float* fu_w2    = (const float*)d_in[34];
    const float* fu_b2    = (const float*)d_in[35];
    const float* fu_w3    = (const float*)d_in[36];
    const float* fu_b3    = (const float*)d_in[37];
    const float* fu_w4    = (const float*)d_in[38];
    const float* fu_b4    = (const float*)d_in[39];
    float* out = (float*)d_out;

    // workspace layout (floats)
    float* ws    = (float*)d_ws;
    float* f     = ws;                         // 512*384
    float* Ha    = f     + (size_t)Bq * FINq;  // 512*256
    float* Hb    = Ha    + (size_t)Bq * 256;   // 512*512
    float* Hc    = Hb    + (size_t)Bq * 512;   // 512*256
    float* Kb    = Hc    + (size_t)Bq * 256;   // 2*64*512
    float* Vb    = Kb    + (size_t)Vv * 64 * 512;
    float* Cc    = Vb    + (size_t)Vv * 64 * 512; // 4 (pad 64)
    float* T0    = Cc    + 64;                 // 2*512*64
    float* amap  = T0    + (size_t)Vv * Bq * 64;
    float* H1    = amap  + (size_t)Vv * Bq * 64;  // 2*512*512
    float* H2    = H1    + (size_t)Vv * Bq * 512;
    float* Obuf  = H2    + (size_t)Vv * Bq * 512; // 2*512*64
    float* anext = Obuf  + (size_t)Vv * Bq * 64;

    // ---- Stage C: fusion MLP -> U_B written straight into d_out ----------
    build_f_kernel<<<Bq, 256, 0, stream>>>(BC, sb_w, sb_b, cm_w1, cm_b1,
                                           cm_w2, cm_b2, f);
    gemm_wmma_f32<true, true><<<gemm_grid(Bq, 256), 256, 0, stream>>>(
        f,  fu_w1, fu_b1, Ha, Bq, 256, FINq, FINq, 1, 256, 1, 256, 0, 1, 0);
    gemm_wmma_f32<true, true><<<gemm_grid(Bq, 512), 256, 0, stream>>>(
        Ha, fu_w2, fu_b2, Hb, Bq, 512, 256, 256, 1, 512, 1, 512, 0, 1, 0);
    gemm_wmma_f32<true, true><<<gemm_grid(Bq, 256), 256, 0, stream>>>(
        Hb, fu_w3, fu_b3, Hc, Bq, 256, 512, 512, 1, 256, 1, 256, 0, 1, 0);
    gemm_wmma_f32<true, true><<<gemm_grid(Bq, Vv * NXq), 256, 0, stream>>>(
        Hc, fu_w4, fu_b4, out, Bq, Vv * NXq, 256, 256, 1, Vv * NXq, 1,
        Vv * NXq, 0, 0, 0);

    // ---- Stage A: batch-independent attention collapse -------------------
    for (int v = 0; v < Vv; v++) {
        // K_v = Phi_v^T @ Wk_w_v + Wk_b   (A[m,k] = Phi[v,k,m]: sa_m=1, sa_k=R)
        gemm_wmma_f32<false, true><<<gemm_grid(Rq, Dq), 256, 0, stream>>>(
            Phi + (size_t)v * NXq * Rq, Wk_w + (size_t)v * NXq * Dq,
            Wk_b + (size_t)v * Dq, Kb + (size_t)v * Rq * Dq,
            Rq, Dq, NXq, 1, Rq, Dq, 1, Dq, 0, 0, 0);
        gemm_wmma_f32<false, true><<<gemm_grid(Rq, Dq), 256, 0, stream>>>(
            Phi + (size_t)v * NXq * Rq, Wv_w + (size_t)v * NXq * Dq,
            Wv_b + (size_t)v * Dq, Vb + (size_t)v * Rq * Dq,
            Rq, Dq, NXq, 1, Rq, Dq, 1, Dq, 0, 0, 0);
    }
    consts_kernel<<<Vv, 512, 0, stream>>>(Kb, Vb, Wq_w, Wq_b, attn_ow,
                                          attn_ob, proj_w, proj_b, Cc);

    // ---- Stage B: per-batch FFN chain -------------------------------------
    for (int v = 0; v < Vv; v++) {
        // T0 = relu(a_n_v @ a0_w_v + a0_b_v); a_n rows strided by V*R
        gemm_wmma_f32<true, true><<<gemm_grid(Bq, Rq), 256, 0, stream>>>(
            a_n + (size_t)v * Rq, a0_w + (size_t)v * Rq * Rq,
            a0_b + (size_t)v * Rq, T0 + (size_t)v * Bq * Rq,
            Bq, Rq, Rq, Vv * Rq, 1, Rq, 1, Rq, 0, 2, 0);
    }
    ln64_kernel<<<dim3(Bq, Vv), 64, 0, stream>>>(T0, a0_ln_g, a0_ln_b, amap);
    for (int v = 0; v < Vv; v++) {
        gemm_wmma_f32<true, true><<<gemm_grid(Bq, Dq), 256, 0, stream>>>(
            amap + (size_t)v * Bq * Rq, ffn_w1 + (size_t)v * Rq * Dq,
            ffn_b1 + (size_t)v * Dq, H1 + (size_t)v * Bq * Dq,
            Bq, Dq, Rq, Rq, 1, Dq, 1, Dq, 0, 1, 0);
        gemm_wmma_f32<true, true><<<gemm_grid(Bq, Dq), 256, 0, stream>>>(
            H1 + (size_t)v * Bq * Dq, ffn_w2 + (size_t)v * Dq * Dq,
            ffn_b2 + (size_t)v * Dq, H2 + (size_t)v * Bq * Dq,
            Bq, Dq, Dq, Dq, 1, Dq, 1, Dq, 0, 1, 0);
        gemm_wmma_f32<true, true><<<gemm_grid(Bq, Rq), 256, 0, stream>>>(
            H2 + (size_t)v * Bq * Dq, ffn_w3 + (size_t)v * Dq * Rq,
            ffn_b3 + (size_t)v * Rq, Obuf + (size_t)v * Bq * Rq,
            Bq, Rq, Dq, Dq, 1, Rq, 1, Rq, 0, 0, 0);
    }
    anext_kernel<<<dim3(Bq, Vv), 64, 0, stream>>>(Obuf, amap, ffn_ln_g,
                                                  ffn_ln_b, Cc, alpha, anext);

    // ---- Final: out[b, v, n] += a_next_v @ Phi_v^T  (W[k,n]=Phi[v,n,k]) ---
    for (int v = 0; v < Vv; v++) {
        gemm_wmma_f32<true, false><<<gemm_grid(Bq, NXq), 256, 0, stream>>>(
            anext + (size_t)v * Bq * Rq, Phi + (size_t)v * NXq * Rq,
            nullptr, out, Bq, NXq, Rq, Rq, 1, 1, Rq,
            Vv * NXq, v * NXq, 0, 1);
    }
}